// VQC_Net_76287209111807
// MI455X (gfx1250) — compile-verified
//
#include <hip/hip_runtime.h>
#include <math.h>

typedef __attribute__((ext_vector_type(2))) float v2f;
typedef __attribute__((ext_vector_type(8))) float v8f;

#define EPSQ 1e-8f

// fast hardware sqrt: single v_sqrt_f32 (~1 ulp), no IEEE denorm fixup chain
__device__ __forceinline__ float fsqrt(float v) {
    return __builtin_amdgcn_sqrtf(v);
}

// R = [[c,-s],[s,c]]; element [br][bc]
__device__ __forceinline__ float ry_elem(float c, float s, int br, int bc) {
    return br == 0 ? (bc == 0 ? c : -s) : (bc == 0 ? s : c);
}

// Build fused circuit matrix Mc = RY(theta[4:8]) * diag(CZ) * RY(theta[0:4]).
// Qubit q lives on bit q (LSB-first), matching the reference's kron order.
__global__ void build_mc_kernel(const float* __restrict__ theta,
                                float* __restrict__ mc) {
    int t = threadIdx.x;
    int i = t >> 4;   // output row
    int j = t & 15;   // input column
    float c1[4], s1[4], c2[4], s2[4];
#pragma unroll
    for (int q = 0; q < 4; ++q) {
        c1[q] = cosf(theta[q] * 0.5f);
        s1[q] = sinf(theta[q] * 0.5f);
        c2[q] = cosf(theta[4 + q] * 0.5f);
        s2[q] = sinf(theta[4 + q] * 0.5f);
    }
    float sum = 0.f;
#pragma unroll
    for (int k = 0; k < 16; ++k) {
        float m2 = 1.f, m1 = 1.f;
#pragma unroll
        for (int q = 0; q < 4; ++q) {
            m2 *= ry_elem(c2[q], s2[q], (i >> q) & 1, (k >> q) & 1);
            m1 *= ry_elem(c1[q], s1[q], (k >> q) & 1, (j >> q) & 1);
        }
        // cz(2,3)*cz(1,2)*cz(0,1)*cz(0,3) folded sign
        int neg = (((k >> 2) & (k >> 3)) ^ ((k >> 1) & (k >> 2)) ^
                   ((k >> 0) & (k >> 1)) ^ ((k >> 0) & (k >> 3))) & 1;
        sum += (neg ? -m2 : m2) * m1;
    }
    mc[i * 16 + j] = sum;
}

// Each wave: 16 batch columns per tile, 8 tiles. state_out = Mc * state_in via
// 4x chained V_WMMA_F32_16X16X4_F32, then square + per-qubit bit-mask reduce.
__global__ void __launch_bounds__(256) vqc_main_kernel(
        const float* __restrict__ x, const float* __restrict__ mc,
        float* __restrict__ out, int B) {
    const int lane = threadIdx.x & 31;
    const int half = lane >> 4;   // 0: K rows 4c..4c+1 / D rows 0-7; 1: K rows 4c+2..3 / D rows 8-15
    const int col  = lane & 15;   // batch column within tile / matrix row for A
    const long wave = (long)((blockIdx.x * blockDim.x + threadIdx.x) >> 5);
    const long waveBase = wave * 128;
    if (waveBase >= B) return;

    // A-matrix chunks of Mc (16x4 each). ISA layout: lanes0-15 M=row, lanes16-31 M=row;
    // v0 = K=(2*half), v1 = K=(2*half+1) within the chunk.
    const float* mrow = mc + col * 16 + 2 * half;
    v2f a0 = *(const v2f*)(mrow + 0);
    v2f a1 = *(const v2f*)(mrow + 4);
    v2f a2 = *(const v2f*)(mrow + 8);
    v2f a3 = *(const v2f*)(mrow + 12);

    for (int t = 0; t < 8; ++t) {
        long base = waveBase + (long)t * 16;
        if (base >= B) break;                  // wave-uniform
        long b  = base + col;
        long bl = b < B ? b : (long)(B - 1);   // clamp loads on ragged tail
        float4 xv = *(const float4*)(x + bl * 4);
        // y[q] = x[b][3-q]  (reference BITS are MSB-first)
        float y0 = xv.w, y1 = xv.z, y2 = xv.y, y3 = xv.x;

        // B-matrix rows handled by this lane in chunk c: r = 4c + 2*half (+1).
        // bit0(r)=0/1 selects (1-y0)/y0; bit1(r)=half; bit2=c&1; bit3=c>>1.
        float f1 = half ? y1 : (1.f - y1);
        float e0 = 1.f - y0;
        float n2 = 1.f - y2, n3 = 1.f - y3;

        v8f acc = {0.f, 0.f, 0.f, 0.f, 0.f, 0.f, 0.f, 0.f};
        {   // chunk 0: bit2=0, bit3=0
            float p = f1 * n2 * n3;
            v2f bm; bm.x = fsqrt(p * e0 + EPSQ); bm.y = fsqrt(p * y0 + EPSQ);
            acc = __builtin_amdgcn_wmma_f32_16x16x4_f32(false, a0, false, bm,
                                                        (short)0, acc, false, false);
        }
        {   // chunk 1: bit2=1, bit3=0
            float p = f1 * y2 * n3;
            v2f bm; bm.x = fsqrt(p * e0 + EPSQ); bm.y = fsqrt(p * y0 + EPSQ);
            acc = __builtin_amdgcn_wmma_f32_16x16x4_f32(false, a1, false, bm,
                                                        (short)0, acc, false, false);
        }
        {   // chunk 2: bit2=0, bit3=1
            float p = f1 * n2 * y3;
            v2f bm; bm.x = fsqrt(p * e0 + EPSQ); bm.y = fsqrt(p * y0 + EPSQ);
            acc = __builtin_amdgcn_wmma_f32_16x16x4_f32(false, a2, false, bm,
                                                        (short)0, acc, false, false);
        }
        {   // chunk 3: bit2=1, bit3=1
            float p = f1 * y2 * y3;
            v2f bm; bm.x = fsqrt(p * e0 + EPSQ); bm.y = fsqrt(p * y0 + EPSQ);
            acc = __builtin_amdgcn_wmma_f32_16x16x4_f32(false, a3, false, bm,
                                                        (short)0, acc, false, false);
        }

        // D layout: this lane's column = col; rows v + 8*half for v in 0..7.
        float pv[8];
#pragma unroll
        for (int v = 0; v < 8; ++v) { float d = acc[v]; pv[v] = d * d; }
        float S0 = pv[1] + pv[3] + pv[5] + pv[7];   // qubit 0: row bit0 = v&1
        float S1 = pv[2] + pv[3] + pv[6] + pv[7];   // qubit 1
        float S2 = pv[4] + pv[5] + pv[6] + pv[7];   // qubit 2
        float Sall = S2 + pv[0] + pv[1] + pv[2] + pv[3];
        float S3 = half ? Sall : 0.f;               // qubit 3: row bit3 = half

        // combine the two half-wave partial sums for each column
        S0 += __shfl_xor(S0, 16, 32);
        S1 += __shfl_xor(S1, 16, 32);
        S2 += __shfl_xor(S2, 16, 32);
        S3 += __shfl_xor(S3, 16, 32);

        if (half == 0 && b < B) {
            // out[b][j] = E[bit (3-j)]  ->  {S3, S2, S1, S0}
            float4 o = make_float4(S3, S2, S1, S0);
            *(float4*)(out + b * 4) = o;
        }
    }
}

extern "C" void kernel_launch(void* const* d_in, const int* in_sizes, int n_in,
                              void* d_out, int out_size, void* d_ws, size_t ws_size,
                              hipStream_t stream) {
    const float* x     = (const float*)d_in[0];
    const float* theta = (const float*)d_in[1];
    float* out = (float*)d_out;
    float* mc  = (float*)d_ws;                 // 256 floats = 1 KB
    int B = in_sizes[0] / 4;
    if (B <= 0) return;

    build_mc_kernel<<<1, 256, 0, stream>>>(theta, mc);

    long nWaves  = ((long)B + 127) / 128;      // 128 batch elems per wave
    long nBlocks = (nWaves + 7) / 8;           // 8 waves per 256-thread block
    vqc_main_kernel<<<(int)nBlocks, 256, 0, stream>>>(x, mc, out, B);
}